// starConvexLoss_12232066859596
// MI455X (gfx1250) — compile-verified
//
#include <hip/hip_runtime.h>
#include <hip/hip_bf16.h>

// Problem constants (match reference)
#define B_ROWS   65536
#define C_CLS    10
#define S_NOISE  128
#define NB       256              // rows per chunk
#define CHUNKS   256              // B_ROWS / NB
#define SGROUPS  4
#define S_PER_G  32               // S_NOISE / SGROUPS
#define TILE_FLOATS (NB * C_CLS)  // 2560 floats = 10240 bytes per (s,chunk) tile
#define NPART    (S_NOISE * CHUNKS)

typedef float v2f __attribute__((ext_vector_type(2)));
typedef float v8f __attribute__((ext_vector_type(8)));

__device__ __forceinline__ float wave_sum(float v) {
#pragma unroll
  for (int o = 16; o > 0; o >>= 1) v += __shfl_down(v, o, 32);
  return v;
}

// Stream one 10240-byte tile (NB rows x 10 floats) of noise into LDS with
// CDNA5 async-to-LDS loads. 1280 b64 units / 256 threads = 5 per thread
// (uniform per wave, so ASYNCcnt bookkeeping is exact).
__device__ __forceinline__ void async_tile(const float* gbase, unsigned ldsb) {
  const unsigned t = threadIdx.x;
#pragma unroll
  for (int k = 0; k < 5; ++k) {
    unsigned off = (t + (unsigned)k * 256u) * 8u;
    asm volatile("global_load_async_to_lds_b64 %0, %1, %2 offset:0"
                 :: "v"(ldsb + off), "v"(off), "s"(gbase)
                 : "memory");
  }
}

__global__ __launch_bounds__(256)
void sc_main(const float* __restrict__ pred,
             const int*   __restrict__ tgt,
             const float* __restrict__ noise,
             float* __restrict__ P,        // [3][S_NOISE][CHUNKS]
             float* __restrict__ cePart)   // [CHUNKS]
{
  __shared__ __align__(16) float nbuf[2][TILE_FLOATS];
  __shared__ float redbuf[8][4];

  const int t     = threadIdx.x;
  const int chunk = blockIdx.x & (CHUNKS - 1);
  const int sg    = blockIdx.x >> 8;        // / CHUNKS
  const int s0    = sg * S_PER_G;
  const int b     = chunk * NB + t;         // this thread's row for all phases

  const unsigned lb0 = (unsigned)(size_t)(&nbuf[0][0]);  // flat addr low 32 = LDS offset
  const unsigned lb1 = (unsigned)(size_t)(&nbuf[1][0]);

  // Kick off the first noise tile before doing any row math (overlap).
  {
    const float* g0 = noise + ((size_t)s0 * B_ROWS + (size_t)chunk * NB) * C_CLS;
    async_tile(g0, lb0);
  }

  // ---- Phase A: per-row prediction stats (L2-resident, recomputed) ----
  float p[C_CLS];
#pragma unroll
  for (int c = 0; c < C_CLS; ++c) p[c] = pred[(size_t)b * C_CLS + c];
  const int tg = tgt[b];

  float mx = p[0];
#pragma unroll
  for (int c = 1; c < C_CLS; ++c) mx = fmaxf(mx, p[c]);
  float se = 0.0f;
#pragma unroll
  for (int c = 0; c < C_CLS; ++c) se += __expf(p[c] - mx);
  const float K = mx + __logf(se);          // logsumexp; logp[c] = p[c] - K

  if (sg == 0) {                            // hard-CE partials, computed once
    float ptg = 0.0f;
#pragma unroll
    for (int c = 0; c < C_CLS; ++c) ptg = (c == tg) ? p[c] : ptg;
    float ce = wave_sum(K - ptg);
    if ((t & 31) == 0) redbuf[t >> 5][0] = ce;
    __syncthreads();
    if (t == 0) {
      float s = 0.0f;
      for (int i = 0; i < 8; ++i) s += redbuf[i][0];
      cePart[chunk] = s;
    }
    __syncthreads();
  }

  // ---- Phase B: stream noise tiles, double-buffered via ASYNCcnt ----
  for (int s = s0; s < s0 + S_PER_G; ++s) {
    if (s + 1 < s0 + S_PER_G) {
      const float* gn = noise + ((size_t)(s + 1) * B_ROWS + (size_t)chunk * NB) * C_CLS;
      async_tile(gn, ((s + 1) & 1) ? lb1 : lb0);
      asm volatile("s_wait_asynccnt 5" ::: "memory");   // current tile's 5 done
    } else {
      asm volatile("s_wait_asynccnt 0" ::: "memory");
    }
    __syncthreads();                                    // whole tile visible

    const float* nb = (s & 1) ? &nbuf[1][0] : &nbuf[0][0];
    float nz[C_CLS];
#pragma unroll
    for (int c = 0; c < C_CLS; ++c) nz[c] = nb[t * C_CLS + c];

    float n2 = 0.0f, m1 = -3.4e38f, m2 = -3.4e38f;
#pragma unroll
    for (int c = 0; c < C_CLS; ++c) {
      const float oh = (c == tg) ? 1.0f : 0.0f;
      m1 = fmaxf(m1, nz[c] + oh);
      m2 = fmaxf(m2, 0.5f * nz[c] + oh);
      n2 += nz[c] * nz[c];
    }
    float s1 = 0.0f, d1 = 0.0f, s2 = 0.0f, d2 = 0.0f;
#pragma unroll
    for (int c = 0; c < C_CLS; ++c) {
      const float oh = (c == tg) ? 1.0f : 0.0f;
      const float e1 = __expf(nz[c] + oh - m1);
      const float e2 = __expf(0.5f * nz[c] + oh - m2);
      s1 += e1; d1 += e1 * p[c];
      s2 += e2; d2 += e2 * p[c];
    }
    // dot(softmax(z), logp_row) = dot(e,p)/sum(e) - K   (softmax sums to 1)
    float dw  = wave_sum(d1 / s1 - K);
    float dwt = wave_sum(d2 / s2 - K);
    float dn2 = wave_sum(n2);
    if ((t & 31) == 0) {
      const int w = t >> 5;
      redbuf[w][0] = dw; redbuf[w][1] = dwt; redbuf[w][2] = dn2;
    }
    __syncthreads();                                    // also guards buffer reuse
    if (t == 0) {
      float a0 = 0.0f, a1 = 0.0f, a2 = 0.0f;
      for (int i = 0; i < 8; ++i) { a0 += redbuf[i][0]; a1 += redbuf[i][1]; a2 += redbuf[i][2]; }
      const size_t idx = (size_t)s * CHUNKS + chunk;
      P[idx] = a0; P[NPART + idx] = a1; P[2 * (size_t)NPART + idx] = a2;
    }
  }
}

// Final reduction: sum the 256 chunk-partials for each (quantity, s) with
// V_WMMA_F32_16X16X4_F32 (A = 16 s-rows x 4 chunks, B = ones), then fold the
// 384 ReLU terms into the scalar loss.
__global__ __launch_bounds__(256)
void sc_final(const float* __restrict__ P, const float* __restrict__ cePart,
              float* __restrict__ out)
{
  __shared__ float red[3][S_NOISE];
  __shared__ float ceRed[8];

  const int t    = threadIdx.x;
  const int w    = t >> 5;        // wave id = s-tile id (8 tiles of 16 s)
  const int lane = t & 31;

  float ce = wave_sum(cePart[t]);
  if (lane == 0) ceRed[w] = ce;

  const int m  = lane & 15;             // A-matrix row (s within tile)
  const int kk = (lane >> 4) << 1;      // K pair held by this half-wave
  const v2f ones = {1.0f, 1.0f};

  for (int q = 0; q < 3; ++q) {
    const float* Pr = P + (size_t)q * NPART + (size_t)(w * 16 + m) * CHUNKS;
    v8f acc = {0.f, 0.f, 0.f, 0.f, 0.f, 0.f, 0.f, 0.f};
    for (int kb = 0; kb < CHUNKS; kb += 4) {
      v2f a = {Pr[kb + kk], Pr[kb + kk + 1]};
      acc = __builtin_amdgcn_wmma_f32_16x16x4_f32(
          /*neg_a=*/false, a, /*neg_b=*/false, ones,
          /*c_mod=*/(short)0, acc, /*reuse_a=*/false, /*reuse_b=*/false);
    }
    // D rows replicate the chunk-sum across all 16 columns.
    if (lane == 0) {
#pragma unroll
      for (int r = 0; r < 8; ++r) red[q][w * 16 + r] = acc[r];
    } else if (lane == 16) {
#pragma unroll
      for (int r = 0; r < 8; ++r) red[q][w * 16 + 8 + r] = acc[r];
    }
  }
  __syncthreads();

  if (t == 0) {
    float ces = 0.0f;
    for (int i = 0; i < 8; ++i) ces += ceRed[i];
    const float invB  = 1.0f / (float)B_ROWS;
    const float wstar = ces * invB;
    float sc = 0.0f;
    for (int s = 0; s < S_NOISE; ++s) {
      const float wv  = -red[0][s] * invB;
      const float wtv = -red[1][s] * invB;
      const float n2  =  red[2][s];
      const float sc1 = fmaxf(wstar - wtv, 0.0f);
      const float sc2 = fmaxf(wstar - wv + 0.05f * n2, 0.0f);          // MU/2
      const float sc3 = fmaxf(wtv - 0.5f * wstar + 0.5f * wv
                              + 0.0125f * n2, 0.0f);                    // MU*LAM*(1-LAM)/2
      sc += sc1 + sc2 + sc3;
    }
    out[0] = wstar + 0.05f * sc;                                        // RHO
  }
}

extern "C" void kernel_launch(void* const* d_in, const int* in_sizes, int n_in,
                              void* d_out, int out_size, void* d_ws, size_t ws_size,
                              hipStream_t stream) {
  (void)in_sizes; (void)n_in; (void)out_size; (void)ws_size;
  const float* pred  = (const float*)d_in[0];
  const int*   tgt   = (const int*)d_in[1];
  const float* noise = (const float*)d_in[2];
  float* P      = (float*)d_ws;              // 3 * 128 * 256 floats
  float* cePart = P + 3 * (size_t)NPART;     // + 256 floats (~394 KB total)

  sc_main<<<dim3(CHUNKS * SGROUPS), dim3(256), 0, stream>>>(pred, tgt, noise, P, cePart);
  sc_final<<<dim3(1), dim3(256), 0, stream>>>(P, cePart, (float*)d_out);
}